// GAT_Variant_5463198401301
// MI455X (gfx1250) — compile-verified
//
#include <hip/hip_runtime.h>
#include <hip/hip_bf16.h>

typedef __attribute__((ext_vector_type(2))) float v2f;
typedef __attribute__((ext_vector_type(8))) float v8f;

#define HEADS 4
#define HC 256           // heads * channels (all 3 layers output 256)
#define CPH 64           // channels per head
#define NEG_SLOPE 0.2f

// ---------------- order-preserving float<->uint encoding for atomicMax ----
__device__ __forceinline__ unsigned enc_f32(float f) {
    unsigned u = __float_as_uint(f);
    return u ^ ((u >> 31) ? 0xFFFFFFFFu : 0x80000000u);
}
__device__ __forceinline__ float dec_f32(unsigned u) {
    return __uint_as_float(u ^ ((u >> 31) ? 0x80000000u : 0xFFFFFFFFu));
}

__device__ __forceinline__ void edge_sd(const int* __restrict__ ei, int E, int e,
                                        int& s, int& d) {
    if (e < E) { s = ei[e]; d = ei[E + e]; }
    else       { s = d = e - E; }           // appended self-loops
}

// ---------------- GEMM: H[M,256] = X[M,K] @ W[K,256] via V_WMMA_F32_16X16X4_F32
// One wave computes a 16x64 output strip (4 accumulators). The A-fragment is
// loaded once per K-step and reused across 4 WMMAs (4 B-fragments), so the
// loop body is 4 wmma : 5 b64-loads instead of 1 : 2.
//   A 16x4 : lanes 0-15 -> M=lane, {K=kb, kb+1}; lanes 16-31 -> {K=kb+2, kb+3}
//   B 4x16 : lanes 0-15 -> N=lane, rows {kb, kb+1}; lanes 16-31 rows {kb+2, kb+3}
//   C/D    : vgpr r -> (M = r + 8*half, N = lane&15)
__global__ __launch_bounds__(256) void gemm_wmma_f32(
        const float* __restrict__ X, const float* __restrict__ W,
        float* __restrict__ H, int M, int K) {
    const int wavesPerBlock = 8;                     // 256 threads / wave32
    int wave  = blockIdx.x * wavesPerBlock + (threadIdx.x >> 5);
    const int strips = HC / 64;                      // 4 strips of 64 cols
    int tileM = wave / strips;
    int strip = wave % strips;
    if (tileM * 16 >= M) return;

    int lane = threadIdx.x & 31;
    int half = lane >> 4;
    int l16  = lane & 15;
    int m    = tileM * 16 + l16;
    int col0 = strip * 64 + l16;                     // cols col0, +16, +32, +48
    const float* xrow = X + (size_t)m * K;

    v8f acc0 = {}, acc1 = {}, acc2 = {}, acc3 = {};
    for (int kb = 0; kb < K; kb += 4) {
        int ka = kb + 2 * half;
        v2f a;
        a.x = xrow[ka];
        a.y = xrow[ka + 1];
        const float* w0 = W + (size_t)ka * HC + col0;
        const float* w1 = W + (size_t)(ka + 1) * HC + col0;
        v2f b0, b1, b2, b3;
        b0.x = w0[0];  b0.y = w1[0];
        b1.x = w0[16]; b1.y = w1[16];
        b2.x = w0[32]; b2.y = w1[32];
        b3.x = w0[48]; b3.y = w1[48];
        acc0 = __builtin_amdgcn_wmma_f32_16x16x4_f32(
                false, a, false, b0, (short)0, acc0, false, false);
        acc1 = __builtin_amdgcn_wmma_f32_16x16x4_f32(
                false, a, false, b1, (short)0, acc1, false, false);
        acc2 = __builtin_amdgcn_wmma_f32_16x16x4_f32(
                false, a, false, b2, (short)0, acc2, false, false);
        acc3 = __builtin_amdgcn_wmma_f32_16x16x4_f32(
                false, a, false, b3, (short)0, acc3, false, false);
    }
#pragma unroll
    for (int r = 0; r < 8; ++r) {
        int row = tileM * 16 + r + 8 * half;
        if (row < M) {
            float* hrow = H + (size_t)row * HC + col0;
            hrow[0]  = acc0[r];
            hrow[16] = acc1[r];
            hrow[32] = acc2[r];
            hrow[48] = acc3[r];
        }
    }
}

// ---------------- per-(node,head) attention scalars: alpha = <h[n,h,:], a[h,:]>
__global__ void node_alpha(const float* __restrict__ H,
                           const float* __restrict__ a_src,
                           const float* __restrict__ a_dst,
                           float* __restrict__ alphaS,
                           float* __restrict__ alphaD, int N) {
    int t = blockIdx.x * blockDim.x + threadIdx.x;
    if (t >= N * HEADS) return;
    int n = t >> 2, h = t & 3;
    const float4* hp = (const float4*)(H + (size_t)n * HC + h * CPH);
    const float4* as = (const float4*)(a_src + h * CPH);
    const float4* ad = (const float4*)(a_dst + h * CPH);
    float ss = 0.f, sd = 0.f;
#pragma unroll 4
    for (int i = 0; i < CPH / 4; ++i) {
        float4 hv = hp[i], av = as[i], dv = ad[i];
        ss += hv.x * av.x + hv.y * av.y + hv.z * av.z + hv.w * av.w;
        sd += hv.x * dv.x + hv.y * dv.y + hv.z * dv.z + hv.w * dv.w;
    }
    alphaS[t] = ss;
    alphaD[t] = sd;
}

// ---------------- fills ----------------
__global__ void fill_f32(float* __restrict__ p, float v, size_t n) {
    size_t i = (size_t)blockIdx.x * blockDim.x + threadIdx.x;
    if (i < n) p[i] = v;
}
__global__ void fill_u32(unsigned* __restrict__ p, unsigned v, size_t n) {
    size_t i = (size_t)blockIdx.x * blockDim.x + threadIdx.x;
    if (i < n) p[i] = v;
}

// ---------------- edge pass 1: segment max of leaky-relu logits ----------
__global__ void edge_max(const int* __restrict__ ei, int E, int N,
                         const float* __restrict__ aS,
                         const float* __restrict__ aD,
                         unsigned* __restrict__ mEnc) {
    int e = blockIdx.x * blockDim.x + threadIdx.x;
    if (e >= E + N) return;
    int s, d; edge_sd(ei, E, e, s, d);
    float4 as = *(const float4*)(aS + 4 * (size_t)s);
    float4 ad = *(const float4*)(aD + 4 * (size_t)d);
    float v[4] = {as.x + ad.x, as.y + ad.y, as.z + ad.z, as.w + ad.w};
#pragma unroll
    for (int h = 0; h < 4; ++h) {
        float x = v[h];
        x = x >= 0.f ? x : NEG_SLOPE * x;
        atomicMax(&mEnc[4 * (size_t)d + h], enc_f32(x));
    }
}

// ---------------- edge pass 2: segment sum of exp(e - max) --------------
__global__ void edge_sum(const int* __restrict__ ei, int E, int N,
                         const float* __restrict__ aS,
                         const float* __restrict__ aD,
                         const unsigned* __restrict__ mEnc,
                         float* __restrict__ sums) {
    int e = blockIdx.x * blockDim.x + threadIdx.x;
    if (e >= E + N) return;
    int s, d; edge_sd(ei, E, e, s, d);
    float4 as = *(const float4*)(aS + 4 * (size_t)s);
    float4 ad = *(const float4*)(aD + 4 * (size_t)d);
    float v[4] = {as.x + ad.x, as.y + ad.y, as.z + ad.z, as.w + ad.w};
#pragma unroll
    for (int h = 0; h < 4; ++h) {
        float x = v[h];
        x = x >= 0.f ? x : NEG_SLOPE * x;
        float m = dec_f32(mEnc[4 * (size_t)d + h]);
        atomicAdd(&sums[4 * (size_t)d + h], expf(x - m));
    }
}

// ---------------- edge pass 3: weighted scatter-add of messages ---------
// one thread per (edge, quad-of-features): 64 threads/edge, float4 gather,
// 4x fp32 atomic adds into the destination row (stays resident in L2).
__global__ void edge_aggr(const int* __restrict__ ei, int E, int N,
                          const float* __restrict__ H,
                          const float* __restrict__ aS,
                          const float* __restrict__ aD,
                          const unsigned* __restrict__ mEnc,
                          const float* __restrict__ sums,
                          float* __restrict__ agg) {
    long long t = (long long)blockIdx.x * blockDim.x + threadIdx.x;
    long long Et = (long long)E + N;
    if (t >= Et * 64) return;
    int e  = (int)(t >> 6);
    int fq = (int)(t & 63);           // quad index: features [4*fq, 4*fq+3]
    int h  = fq >> 4;                 // 16 quads per head
    int s, d; edge_sd(ei, E, e, s, d);
    float x = aS[4 * (size_t)s + h] + aD[4 * (size_t)d + h];
    x = x >= 0.f ? x : NEG_SLOPE * x;
    float m     = dec_f32(mEnc[4 * (size_t)d + h]);
    float denom = sums[4 * (size_t)d + h] + 1e-16f;
    float alpha = expf(x - m) / denom;
    float4 hv = *(const float4*)(H + (size_t)s * HC + fq * 4);
    float* o  = agg + (size_t)d * HC + fq * 4;
    atomicAdd(o + 0, hv.x * alpha);
    atomicAdd(o + 1, hv.y * alpha);
    atomicAdd(o + 2, hv.z * alpha);
    atomicAdd(o + 3, hv.w * alpha);
}

// ---------------- bias (+ optional ReLU), float4 wide -------------------
__global__ void finalize_bias(float* __restrict__ agg,
                              const float* __restrict__ b, int N, int relu) {
    int t = blockIdx.x * blockDim.x + threadIdx.x;
    if (t >= N * (HC / 4)) return;
    int n = t / (HC / 4), fq = t % (HC / 4);
    float4* p = (float4*)(agg + (size_t)n * HC + fq * 4);
    float4 v  = *p;
    float4 bb = *(const float4*)(b + fq * 4);
    v.x += bb.x; v.y += bb.y; v.z += bb.z; v.w += bb.w;
    if (relu) {
        v.x = fmaxf(v.x, 0.f); v.y = fmaxf(v.y, 0.f);
        v.z = fmaxf(v.z, 0.f); v.w = fmaxf(v.w, 0.f);
    }
    *p = v;
}

// ---------------- one GAT layer -----------------------------------------
static void gat_layer(const float* Xin, int K, const float* W, const float* as,
                      const float* ad, const float* b, float* Hbuf, float* agg,
                      const int* ei, int E, int N, float* alphaS, float* alphaD,
                      unsigned* mEnc, float* sums, int relu, hipStream_t stream) {
    int waves  = ((N + 15) / 16) * (HC / 64);          // 16x64 strip per wave
    int blocks = (waves + 7) / 8;                      // 8 waves / block
    gemm_wmma_f32<<<blocks, 256, 0, stream>>>(Xin, W, Hbuf, N, K);

    int nh = N * HEADS;
    node_alpha<<<(nh + 255) / 256, 256, 0, stream>>>(Hbuf, as, ad, alphaS, alphaD, N);
    fill_u32<<<(nh + 255) / 256, 256, 0, stream>>>(mEnc, 0u, (size_t)nh);
    fill_f32<<<(nh + 255) / 256, 256, 0, stream>>>(sums, 0.f, (size_t)nh);
    size_t nagg = (size_t)N * HC;
    fill_f32<<<(int)((nagg + 255) / 256), 256, 0, stream>>>(agg, 0.f, nagg);

    int Et = E + N;
    edge_max<<<(Et + 255) / 256, 256, 0, stream>>>(ei, E, N, alphaS, alphaD, mEnc);
    edge_sum<<<(Et + 255) / 256, 256, 0, stream>>>(ei, E, N, alphaS, alphaD, mEnc, sums);
    long long tt = (long long)Et * 64;
    edge_aggr<<<(int)((tt + 255) / 256), 256, 0, stream>>>(ei, E, N, Hbuf, alphaS,
                                                           alphaD, mEnc, sums, agg);
    finalize_bias<<<(N * (HC / 4) + 255) / 256, 256, 0, stream>>>(agg, b, N, relu);
}

extern "C" void kernel_launch(void* const* d_in, const int* in_sizes, int n_in,
                              void* d_out, int out_size, void* d_ws, size_t ws_size,
                              hipStream_t stream) {
    const float* x   = (const float*)d_in[0];
    const int*   ei  = (const int*)d_in[1];
    const float* W1  = (const float*)d_in[2];
    const float* as1 = (const float*)d_in[3];
    const float* ad1 = (const float*)d_in[4];
    const float* b1  = (const float*)d_in[5];
    const float* W2  = (const float*)d_in[6];
    const float* as2 = (const float*)d_in[7];
    const float* ad2 = (const float*)d_in[8];
    const float* b2  = (const float*)d_in[9];
    const float* W3  = (const float*)d_in[10];
    const float* as3 = (const float*)d_in[11];
    const float* ad3 = (const float*)d_in[12];
    const float* b3  = (const float*)d_in[13];

    int K1 = in_sizes[2] / HC;          // 128
    int K2 = in_sizes[6] / HC;          // 256
    int K3 = in_sizes[10] / HC;         // 256
    int N  = in_sizes[0] / K1;          // 50000
    int E  = in_sizes[1] / 2;           // 800000

    // workspace carve-up
    float*    Hbuf   = (float*)d_ws;
    float*    bufA   = Hbuf + (size_t)N * HC;
    float*    bufB   = bufA + (size_t)N * HC;
    float*    alphaS = bufB + (size_t)N * HC;
    float*    alphaD = alphaS + (size_t)N * HEADS;
    unsigned* mEnc   = (unsigned*)(alphaD + (size_t)N * HEADS);
    float*    sums   = (float*)(mEnc + (size_t)N * HEADS);

    gat_layer(x,    K1, W1, as1, ad1, b1, Hbuf, bufA,          ei, E, N,
              alphaS, alphaD, mEnc, sums, /*relu=*/1, stream);
    gat_layer(bufA, K2, W2, as2, ad2, b2, Hbuf, bufB,          ei, E, N,
              alphaS, alphaD, mEnc, sums, /*relu=*/1, stream);
    gat_layer(bufB, K3, W3, as3, ad3, b3, Hbuf, (float*)d_out, ei, E, N,
              alphaS, alphaD, mEnc, sums, /*relu=*/0, stream);
}